// GCNProtein_31327491457491
// MI455X (gfx1250) — compile-verified
//
#include <hip/hip_runtime.h>
#include <hip/hip_bf16.h>

// GCN 6-layer pipeline for MI455X (gfx1250, wave32).
//   Layer: hn = h*norm ; agg[dst] += hn[src] (16M edges, hw f32 atomics in L2) ;
//          h_next = relu((agg*norm) @ W + b) [* norm folded for next layer]
// Dense update: V_WMMA_F32_16X16X4_F32 (exact fp32). D is transposed through
// LDS (stride-17 rows, conflict-free) so global stores are dense/coalesced.

typedef __attribute__((ext_vector_type(2))) float v2f;
typedef __attribute__((ext_vector_type(8))) float v8f;

// ---------------- elementwise helpers ----------------

__global__ void zero_f32_kernel(float* __restrict__ p, int n) {
    int i = blockIdx.x * blockDim.x + threadIdx.x;
    if (i < n) p[i] = 0.0f;
}

__global__ void scale_f1_kernel(const float* __restrict__ feat,
                                const float* __restrict__ norm,
                                float* __restrict__ hn, int n) {
    int i = blockIdx.x * blockDim.x + threadIdx.x;
    if (i < n) hn[i] = feat[i] * norm[i];
}

// ---------------- edge scatter: agg[dst] += hn[src] ----------------

template <int F>
__device__ __forceinline__ void scatter_one(const float* __restrict__ hn,
                                            float* __restrict__ agg,
                                            int s, int d) {
    const float* __restrict__ hp = hn + s * F;
    float*       __restrict__ ap = agg + d * F;
#pragma unroll
    for (int f = 0; f < F; ++f) {
        unsafeAtomicAdd(ap + f, hp[f]);   // global_atomic_add_f32, non-returning
    }
}

// 4 edges per thread via int4 loads of the coalesced edge arrays.
template <int F>
__global__ void edge_scatter4_kernel(const float* __restrict__ hn,
                                     const int4* __restrict__ src4,
                                     const int4* __restrict__ dst4,
                                     float* __restrict__ agg, int nQuads) {
    int t = blockIdx.x * blockDim.x + threadIdx.x;
    if (t >= nQuads) return;
    if (t + 4096 < nQuads) {
        __builtin_prefetch(&src4[t + 4096], 0, 1);   // global_prefetch_b8
        __builtin_prefetch(&dst4[t + 4096], 0, 1);
    }
    int4 s = src4[t];
    int4 d = dst4[t];
    scatter_one<F>(hn, agg, s.x, d.x);
    scatter_one<F>(hn, agg, s.y, d.y);
    scatter_one<F>(hn, agg, s.z, d.z);
    scatter_one<F>(hn, agg, s.w, d.w);
}

// scalar tail for E % 4 edges (generic-E safety; unused when E = 16M)
template <int F>
__global__ void edge_scatter_tail_kernel(const float* __restrict__ hn,
                                         const int* __restrict__ src,
                                         const int* __restrict__ dst,
                                         float* __restrict__ agg,
                                         int first, int nE) {
    int e = first + blockIdx.x * blockDim.x + threadIdx.x;
    if (e >= nE) return;
    scatter_one<F>(hn, agg, src[e], dst[e]);
}

// ---------------- node update via WMMA f32 16x16x4 ----------------
// Wave handles 32 nodes = two 16-node tiles.
// A (16x4, MxK): lanes 0-15 -> M=lane, VGPR0=K0/VGPR1=K1; lanes 16-31 -> K2/K3.
// B (4x16, KxN): VGPR0 = row K0 (lo lanes) / K2 (hi lanes); VGPR1 = K1 / K3.
// D (16x16):     VGPR r -> M=r (lo lanes, N=lane), M=8+r (hi lanes, N=lane-16).
// Writeback: D -> LDS (32 rows x 16 cols, stride 17) -> dense coalesced stores.

#define LROW 17   // LDS row stride (dwords): avoids lo/hi half bank conflicts

template <int F_IN, int F_OUT, bool NORM_OUT>
__global__ void gcn_node_wmma_kernel(const float* __restrict__ agg,
                                     const float* __restrict__ norm,
                                     const float* __restrict__ W,
                                     const float* __restrict__ bias,
                                     float* __restrict__ out, int N) {
    __shared__ float xbuf[8][32 * LROW];          // 8 waves / 256-thread block

    const int tid  = threadIdx.x;
    const int lane = tid & 31;
    const int w    = tid >> 5;
    const int wid  = (blockIdx.x * blockDim.x + tid) >> 5;
    const int base = wid * 32;
    const int col  = lane & 15;
    const int hi   = lane >> 4;          // 0: lanes 0-15, 1: lanes 16-31
    const int k0   = 2 * hi;             // this half's first K row

    // ---- B fragment: B[k][n] = W[k*F_OUT + n], zero-padded. Clamped loads. ----
    const int   wcol = (col < F_OUT) ? col : 0;
    const int   kx   = (k0 < F_IN) ? k0 : 0;
    const int   ky   = (k0 + 1 < F_IN) ? (k0 + 1) : 0;
    const float wxv  = W[kx * F_OUT + wcol];
    const float wyv  = W[ky * F_OUT + wcol];
    v2f bm;
    bm.x = (col < F_OUT && k0 < F_IN)     ? wxv : 0.0f;
    bm.y = (col < F_OUT && k0 + 1 < F_IN) ? wyv : 0.0f;
    const float bv = bias[wcol];

    float* __restrict__ lp = xbuf[w];

#pragma unroll
    for (int g = 0; g < 2; ++g) {
        const int m0 = base + g * 16;

        // ---- A fragment: X[m][k] = agg[m][k]*norm[m]; unconditional clamped loads ----
        const int   anode = m0 + col;
        const int   cn    = (anode < N) ? anode : (N - 1);
        const float nrm   = norm[cn];
        v2f am;
        if (F_IN == 1) {
            const float a0 = agg[cn];
            am.x = hi ? 0.0f : a0 * nrm;         // only K0 is real
            am.y = 0.0f;                         // K1/K3 zero
        } else { // F_IN == 3
            const float ax = agg[cn * 3 + k0];   // K0 (lo) / K2 (hi)
            const float ay = agg[cn * 3 + 1];    // K1, lo lanes only
            am.x = ax * nrm;
            am.y = hi ? 0.0f : ay * nrm;         // K3 half zero
        }

        v8f c = {};
        v8f d = __builtin_amdgcn_wmma_f32_16x16x4_f32(
            /*neg_a=*/false, am, /*neg_b=*/false, bm,
            /*c_mod=*/(short)0, c, /*reuse_a=*/false, /*reuse_b=*/false);

        // ---- unconditional LDS writeback: relu(d[r]+b) into [row][col] ----
#pragma unroll
        for (int r = 0; r < 8; ++r) {
            float v = d[r] + bv;
            v = v > 0.0f ? v : 0.0f;                         // relu
            lp[(g * 16 + r + hi * 8) * LROW + col] = v;      // ds_store_b32
        }
    }

    // ---- dense coalesced global stores: F_OUT values per lane ----
#pragma unroll
    for (int k = 0; k < F_OUT; ++k) {
        const int j    = k * 32 + lane;          // flat index within wave's 32*F_OUT outputs
        const int nl   = j / F_OUT;              // local node (compile-time divisor)
        const int c    = j % F_OUT;
        const int node = base + nl;
        const int sn   = (node < N) ? node : (N - 1);
        float v = lp[nl * LROW + c];             // ds_load_b32
        if (NORM_OUT) v *= norm[sn];             // fold next layer's pre-scale
        if (node < N)
            out[base * F_OUT + j] = v;           // contiguous across lanes
    }
}

// ---------------- launcher ----------------

extern "C" void kernel_launch(void* const* d_in, const int* in_sizes, int n_in,
                              void* d_out, int out_size, void* d_ws, size_t ws_size,
                              hipStream_t stream) {
    // setup_inputs() dict order: feat, norm, src, dst, W1,b1, ..., W6,b6
    const float* feat = (const float*)d_in[0];
    const float* norm = (const float*)d_in[1];
    const int*   src  = (const int*)d_in[2];
    const int*   dst  = (const int*)d_in[3];
    const float* W[6];
    const float* B[6];
    for (int i = 0; i < 6; ++i) {
        W[i] = (const float*)d_in[4 + 2 * i];
        B[i] = (const float*)d_in[5 + 2 * i];
    }
    const int N = in_sizes[0];
    const int E = in_sizes[2];

    float* hn  = (float*)d_ws;              // 3N floats (12 MB)
    float* agg = hn + (size_t)3 * N;        // 3N floats (12 MB)
    float* out = (float*)d_out;

    const int TB = 256;
    const int nBlksN  = (N + TB - 1) / TB;
    const int nBlks3N = (3 * N + TB - 1) / TB;
    const int nQuads  = E / 4;
    const int nTail   = E - 4 * nQuads;
    const int nBlksQ  = (nQuads + TB - 1) / TB;
    const int nWaves  = (N + 31) / 32;
    const int nBlksW  = (nWaves + (TB / 32) - 1) / (TB / 32);
    const int4* src4 = (const int4*)src;
    const int4* dst4 = (const int4*)dst;

    // hn0 = feat * norm   (layer 1 input, F=1)
    scale_f1_kernel<<<nBlksN, TB, 0, stream>>>(feat, norm, hn, N);

    // ---- layer 1: F_in=1 -> F_out=3 ----
    zero_f32_kernel<<<nBlksN, TB, 0, stream>>>(agg, N);
    edge_scatter4_kernel<1><<<nBlksQ, TB, 0, stream>>>(hn, src4, dst4, agg, nQuads);
    if (nTail)
        edge_scatter_tail_kernel<1><<<1, TB, 0, stream>>>(hn, src, dst, agg, 4 * nQuads, E);
    gcn_node_wmma_kernel<1, 3, true><<<nBlksW, TB, 0, stream>>>(agg, norm, W[0], B[0], hn, N);

    // ---- layers 2..5: F_in=3 -> F_out=3 ----
    for (int l = 1; l < 5; ++l) {
        zero_f32_kernel<<<nBlks3N, TB, 0, stream>>>(agg, 3 * N);
        edge_scatter4_kernel<3><<<nBlksQ, TB, 0, stream>>>(hn, src4, dst4, agg, nQuads);
        if (nTail)
            edge_scatter_tail_kernel<3><<<1, TB, 0, stream>>>(hn, src, dst, agg, 4 * nQuads, E);
        gcn_node_wmma_kernel<3, 3, true><<<nBlksW, TB, 0, stream>>>(agg, norm, W[l], B[l], hn, N);
    }

    // ---- layer 6: F_in=3 -> F_out=1, no trailing norm, to d_out ----
    zero_f32_kernel<<<nBlks3N, TB, 0, stream>>>(agg, 3 * N);
    edge_scatter4_kernel<3><<<nBlksQ, TB, 0, stream>>>(hn, src4, dst4, agg, nQuads);
    if (nTail)
        edge_scatter_tail_kernel<3><<<1, TB, 0, stream>>>(hn, src, dst, agg, 4 * nQuads, E);
    gcn_node_wmma_kernel<3, 1, false><<<nBlksW, TB, 0, stream>>>(agg, norm, W[5], B[5], out, N);
}